// YOLOXHead_80461917323412
// MI455X (gfx1250) — compile-verified
//
#include <hip/hip_runtime.h>
#include <hip/hip_bf16.h>

typedef float v2f __attribute__((ext_vector_type(2)));
typedef float v8f __attribute__((ext_vector_type(8)));

#define BB 8
#define NN 8192
#define GG 16
#define CC 80
#define ROW (5 + CC)          // 85
#define TOPK 10
#define CENTER_R 2.5f

// ---------- helpers ----------
__device__ __forceinline__ float sigmoidf_(float x) {
    return 1.0f / (1.0f + expf(-x));
}

// D[n,c] = log(p+eps) - log(1-p+eps), p = sqrt(sigmoid(cls)*sigmoid(obj))
__device__ __forceinline__ float dtermf_(float logit, float sobj) {
    float p = sqrtf(sigmoidf_(logit) * sobj);
    return logf(p + 1e-8f) - logf(1.0f - p + 1e-8f);
}

// total-order float -> uint (ascending uint == ascending float)
__device__ __forceinline__ unsigned ford_(float x) {
    unsigned u = __float_as_uint(x);
    return (u & 0x80000000u) ? ~u : (u | 0x80000000u);
}

// ---------- kernel 1: decode, S[n] = sum_c log1mp, init bestKey ----------
__global__ void k1_precompute(const float* __restrict__ outp,
                              const float* __restrict__ pos,
                              const float* __restrict__ str,
                              float* __restrict__ dbox,
                              float* __restrict__ S,
                              unsigned long long* __restrict__ bestKey) {
    int idx = blockIdx.x * blockDim.x + threadIdx.x;   // over B*N
    if (idx >= BB * NN) return;
    const float* row = outp + (size_t)idx * ROW;
    float s = str[idx];
    float cx = (pos[idx * 2 + 0] + row[0]) * s;
    float cy = (pos[idx * 2 + 1] + row[1]) * s;
    float w  = expf(fminf(row[2], 10.0f)) * s;
    float h  = expf(fminf(row[3], 10.0f)) * s;
    float* db = dbox + (size_t)idx * 4;
    db[0] = cx; db[1] = cy; db[2] = w; db[3] = h;
    float sobj = sigmoidf_(row[4]);
    float acc = 0.0f;
    for (int c = 0; c < CC; ++c) {
        float p = sqrtf(sigmoidf_(row[5 + c]) * sobj);
        acc += logf(1.0f - p + 1e-8f);
    }
    S[idx] = acc;
    bestKey[idx] = ~0ull;
}

// ---------- kernel 2: pairwise cost/iou via V_WMMA_F32_16X16X4_F32 ----------
// one wave per 16-anchor tile; A = one-hot(gt class) built in-register,
// B = D[n,c] computed on the fly (lane layout per CDNA5 ISA 16x4 f32 tables).
// K-loop fully unrolled: 20 static WMMAs per tile, independent transcendental
// chains interleave with the matrix-pipe accumulation.
__global__ void k2_pair(const float* __restrict__ outp,
                        const float* __restrict__ labels,
                        const float* __restrict__ dbox,
                        const float* __restrict__ S,
                        const float* __restrict__ str,
                        float* __restrict__ costA,
                        float* __restrict__ iouA,
                        float* __restrict__ candA) {
    __shared__ int   cls_s[GG];
    __shared__ float gb_s[GG][4];
    int b = blockIdx.y;
    int tid = threadIdx.x;
    if (tid < GG) {
        const float* L = labels + (size_t)(b * GG + tid) * 5;
        cls_s[tid] = (int)L[0];
        gb_s[tid][0] = L[1]; gb_s[tid][1] = L[2];
        gb_s[tid][2] = L[3]; gb_s[tid][3] = L[4];
    }
    __syncthreads();

    int wid  = tid >> 5;
    int lane = tid & 31;
    int tile = blockIdx.x * (blockDim.x >> 5) + wid;   // 0..511
    int n    = tile * 16 + (lane & 15);
    size_t aidx = (size_t)b * NN + n;

    const float* row = outp + aidx * ROW;
    float sobj = sigmoidf_(row[4]);
    float Sn   = S[aidx];
    float sn   = str[aidx];
    const float* db = dbox + aidx * 4;
    float px = db[0], py = db[1], pw = db[2], ph = db[3];

    int hi = (lane >= 16) ? 1 : 0;
    int khalf = hi * 2;
    int cm = cls_s[lane & 15];

    v8f acc = {};
#pragma unroll
    for (int kk = 0; kk < CC; kk += 4) {
        int c0 = kk + khalf;
        v2f a, bv;
        a.x = (cm == c0)     ? 1.0f : 0.0f;
        a.y = (cm == c0 + 1) ? 1.0f : 0.0f;
        bv.x = dtermf_(row[5 + c0], sobj);
        bv.y = dtermf_(row[6 + c0], sobj);
        acc = __builtin_amdgcn_wmma_f32_16x16x4_f32(
            false, a, false, bv, (short)0, acc, false, false);
    }

    float p1x = px - pw * 0.5f, p2x = px + pw * 0.5f;
    float p1y = py - ph * 0.5f, p2y = py + ph * 0.5f;
    float pa = pw * ph;
    float rr = CENTER_R * sn;

#pragma unroll
    for (int r = 0; r < 8; ++r) {
        int g = r + hi * 8;
        float gx = gb_s[g][0], gy = gb_s[g][1], gw = gb_s[g][2], gh = gb_s[g][3];
        float g1x = gx - gw * 0.5f, g2x = gx + gw * 0.5f;
        float g1y = gy - gh * 0.5f, g2y = gy + gh * 0.5f;
        bool in_box = (px >= g1x) && (px <= g2x) && (py >= g1y) && (py <= g2y);
        bool in_ctr = (fabsf(px - gx) <= rr) && (fabsf(py - gy) <= rr);
        float iw = fmaxf(fminf(p2x, g2x) - fmaxf(p1x, g1x), 0.0f);
        float ih = fmaxf(fminf(p2y, g2y) - fmaxf(p1y, g1y), 0.0f);
        float inter = iw * ih;
        float iou = inter / (pa + gw * gh - inter + 1e-16f);
        float iou_cost = -logf(iou + 1e-8f);
        float pair_cls = -(acc[r] + Sn);
        float cost = pair_cls + 3.0f * iou_cost +
                     ((in_box && in_ctr) ? 0.0f : 100000.0f);
        size_t o = ((size_t)b * GG + g) * NN + n;
        costA[o] = cost;
        iouA[o]  = iou;
        candA[o] = (in_box || in_ctr) ? iou : 0.0f;
    }
}

// ---------- kernel 3: per (b,g) top-10 selections, scatter matches ----------
__global__ void k3_select(const float* __restrict__ costA,
                          const float* __restrict__ candA,
                          unsigned long long* __restrict__ bestKey) {
    __shared__ float fbuf[256 * TOPK];
    __shared__ unsigned long long kbuf[256 * TOPK];
    int bg = blockIdx.x;
    int b = bg >> 4, g = bg & 15;
    int tid = threadIdx.x;
    const float* cand = candA + (size_t)bg * NN;
    const float* cost = costA + (size_t)bg * NN;

    // top-10 candidate IoU (descending) -> dyn_k
    float t10[TOPK];
#pragma unroll
    for (int i = 0; i < TOPK; ++i) t10[i] = -1.0f;
    for (int nn = tid; nn < NN; nn += 256) {
        float v = cand[nn];
        if (v > t10[TOPK - 1]) {
            int i = TOPK - 1;
            while (i > 0 && t10[i - 1] < v) { t10[i] = t10[i - 1]; --i; }
            t10[i] = v;
        }
    }
#pragma unroll
    for (int i = 0; i < TOPK; ++i) fbuf[tid * TOPK + i] = t10[i];

    // top-10 smallest (cost, idx) lexicographic
    unsigned long long k10[TOPK];
#pragma unroll
    for (int i = 0; i < TOPK; ++i) k10[i] = ~0ull;
    for (int nn = tid; nn < NN; nn += 256) {
        unsigned long long key =
            ((unsigned long long)ford_(cost[nn]) << 32) | (unsigned)nn;
        if (key < k10[TOPK - 1]) {
            int i = TOPK - 1;
            while (i > 0 && k10[i - 1] > key) { k10[i] = k10[i - 1]; --i; }
            k10[i] = key;
        }
    }
#pragma unroll
    for (int i = 0; i < TOPK; ++i) kbuf[tid * TOPK + i] = k10[i];
    __syncthreads();

    if (tid == 0) {
        float m10[TOPK];
#pragma unroll
        for (int i = 0; i < TOPK; ++i) m10[i] = -1.0f;
        for (int j = 0; j < 256 * TOPK; ++j) {
            float v = fbuf[j];
            if (v > m10[TOPK - 1]) {
                int i = TOPK - 1;
                while (i > 0 && m10[i - 1] < v) { m10[i] = m10[i - 1]; --i; }
                m10[i] = v;
            }
        }
        float sum = 0.0f;
#pragma unroll
        for (int i = 0; i < TOPK; ++i) sum += m10[i];
        int dk = (int)sum;
        if (dk < 1) dk = 1;
        if (dk > NN) dk = NN;

        unsigned long long mk[TOPK];
#pragma unroll
        for (int i = 0; i < TOPK; ++i) mk[i] = ~0ull;
        for (int j = 0; j < 256 * TOPK; ++j) {
            unsigned long long key = kbuf[j];
            if (key < mk[TOPK - 1]) {
                int i = TOPK - 1;
                while (i > 0 && mk[i - 1] > key) { mk[i] = mk[i - 1]; --i; }
                mk[i] = key;
            }
        }
        for (int s = 0; s < dk && s < TOPK; ++s) {
            unsigned long long key = mk[s];
            unsigned n = (unsigned)(key & 0xFFFFFFFFull);
            unsigned long long bk =
                (key & 0xFFFFFFFF00000000ull) | (unsigned long long)g;
            atomicMin(&bestKey[(size_t)b * NN + n], bk);
        }
    }
}

// ---------- kernel 4: per-anchor losses + block partial sums ----------
__global__ void k4_loss(const float* __restrict__ outp,
                        const float* __restrict__ labels,
                        const float* __restrict__ dbox,
                        const float* __restrict__ iouA,
                        const unsigned long long* __restrict__ bestKey,
                        float* __restrict__ partial) {
    __shared__ float s0[256], s1[256], s2[256], s3[256];
    int tid = threadIdx.x;
    int idx = blockIdx.x * 256 + tid;       // over B*N (exactly 256 blocks)
    int b = idx / NN, n = idx % NN;

    unsigned long long key = bestKey[idx];
    const float* row = outp + (size_t)idx * ROW;
    bool fg = (key != ~0ull);
    float t = fg ? 1.0f : 0.0f;

    float x = row[4];
    float p = sigmoidf_(x);
    float ce = fmaxf(x, 0.0f) - x * t + log1pf(expf(-fabsf(x)));
    float p_t = p * t + (1.0f - p) * (1.0f - t);
    float a_t = 0.25f * t + 0.75f * (1.0f - t);
    float om = 1.0f - p_t;
    float lo = a_t * om * om * ce;

    float li = 0.0f, lc = 0.0f, nf = 0.0f;
    if (fg) {
        int g = (int)(key & 0xFFFFFFFFull);
        const float* L = labels + (size_t)(b * GG + g) * 5;
        int gc = (int)L[0];
        float gx = L[1], gy = L[2], gw = L[3], gh = L[4];
        const float* db = dbox + (size_t)idx * 4;
        float p1x = db[0] - db[2] * 0.5f, p2x = db[0] + db[2] * 0.5f;
        float p1y = db[1] - db[3] * 0.5f, p2y = db[1] + db[3] * 0.5f;
        float g1x = gx - gw * 0.5f, g2x = gx + gw * 0.5f;
        float g1y = gy - gh * 0.5f, g2y = gy + gh * 0.5f;
        float iw = fmaxf(fminf(p2x, g2x) - fmaxf(p1x, g1x), 0.0f);
        float ih = fmaxf(fminf(p2y, g2y) - fmaxf(p1y, g1y), 0.0f);
        float inter = iw * ih;
        float uni = (p2x - p1x) * (p2y - p1y) + gw * gh - inter + 1e-16f;
        li = 1.0f - inter / uni;
        float miou = iouA[((size_t)b * GG + g) * NN + n];
        for (int c = 0; c < CC; ++c) {
            float xc = row[5 + c];
            float tc = (c == gc) ? miou : 0.0f;
            lc += fmaxf(xc, 0.0f) - xc * tc + log1pf(expf(-fabsf(xc)));
        }
        nf = 1.0f;
    }

    s0[tid] = li; s1[tid] = lo; s2[tid] = lc; s3[tid] = nf;
    __syncthreads();
    for (int off = 128; off > 0; off >>= 1) {
        if (tid < off) {
            s0[tid] += s0[tid + off];
            s1[tid] += s1[tid + off];
            s2[tid] += s2[tid + off];
            s3[tid] += s3[tid + off];
        }
        __syncthreads();
    }
    if (tid == 0) {
        float* pp = partial + blockIdx.x * 4;
        pp[0] = s0[0]; pp[1] = s1[0]; pp[2] = s2[0]; pp[3] = s3[0];
    }
}

// ---------- kernel 5: deterministic final reduction ----------
__global__ void k5_final(const float* __restrict__ partial,
                         float* __restrict__ out) {
    __shared__ float s0[256], s1[256], s2[256], s3[256];
    int tid = threadIdx.x;
    s0[tid] = partial[tid * 4 + 0];
    s1[tid] = partial[tid * 4 + 1];
    s2[tid] = partial[tid * 4 + 2];
    s3[tid] = partial[tid * 4 + 3];
    __syncthreads();
    for (int off = 128; off > 0; off >>= 1) {
        if (tid < off) {
            s0[tid] += s0[tid + off];
            s1[tid] += s1[tid + off];
            s2[tid] += s2[tid + off];
            s3[tid] += s3[tid + off];
        }
        __syncthreads();
    }
    if (tid == 0) {
        float nf = fmaxf(s3[0], 1.0f);
        float Liou = 5.0f * s0[0] / nf;
        float Lobj = s1[0] / nf;
        float Lcls = s2[0] / nf;
        out[0] = Liou + Lobj + Lcls;
        out[1] = Liou;
        out[2] = Lobj;
        out[3] = Lcls;
    }
}

// ---------- workspace layout (bytes) ----------
#define OFF_DBOX   0ull                         // B*N*4 f32   = 1,048,576
#define OFF_S      0x100000ull                  // B*N f32     =   262,144
#define OFF_COST   0x140000ull                  // B*G*N f32   = 4,194,304
#define OFF_IOU    0x540000ull
#define OFF_CAND   0x940000ull
#define OFF_BKEY   0xD40000ull                  // B*N u64     =   524,288
#define OFF_PART   0xDC0000ull                  // 256*4 f32

extern "C" void kernel_launch(void* const* d_in, const int* in_sizes, int n_in,
                              void* d_out, int out_size, void* d_ws, size_t ws_size,
                              hipStream_t stream) {
    (void)in_sizes; (void)n_in; (void)out_size; (void)ws_size;
    const float* outputs   = (const float*)d_in[0];
    const float* positions = (const float*)d_in[1];
    const float* strides   = (const float*)d_in[2];
    const float* labels    = (const float*)d_in[3];
    float* out = (float*)d_out;

    char* ws = (char*)d_ws;
    float* dbox = (float*)(ws + OFF_DBOX);
    float* S    = (float*)(ws + OFF_S);
    float* costA = (float*)(ws + OFF_COST);
    float* iouA  = (float*)(ws + OFF_IOU);
    float* candA = (float*)(ws + OFF_CAND);
    unsigned long long* bestKey = (unsigned long long*)(ws + OFF_BKEY);
    float* partial = (float*)(ws + OFF_PART);

    k1_precompute<<<dim3((BB * NN) / 256), 256, 0, stream>>>(
        outputs, positions, strides, dbox, S, bestKey);
    k2_pair<<<dim3(64, BB), 256, 0, stream>>>(
        outputs, labels, dbox, S, strides, costA, iouA, candA);
    k3_select<<<dim3(BB * GG), 256, 0, stream>>>(costA, candA, bestKey);
    k4_loss<<<dim3((BB * NN) / 256), 256, 0, stream>>>(
        outputs, labels, dbox, iouA, bestKey, partial);
    k5_final<<<1, 256, 0, stream>>>(partial, out);
}